// GDN_86595130622167
// MI455X (gfx1250) — compile-verified
//
#include <hip/hip_runtime.h>
#include <math.h>

// ---- problem constants (match reference) ----
#define NN     10000
#define BB     8
#define FIN    10
#define DIM    64
#define TOPK   20
#define DMODEL 128
#define DINNER 256
#define EPSF   1e-5f
#define PBLK   125           // partial blocks per batch for classifier attw reduction
#define NODES_PER_BLK 80     // NN / PBLK

typedef float v2f __attribute__((ext_vector_type(2)));
typedef float v8f __attribute__((ext_vector_type(8)));

// =====================================================================
// 1. Row-normalize embeddings  (N x 64)
// =====================================================================
__global__ void normalize_kernel(const float* __restrict__ emb, float* __restrict__ out) {
  int n = blockIdx.x, d = threadIdx.x;           // 64 threads
  __shared__ float red[64];
  float v = emb[(size_t)n * DIM + d];
  red[d] = v * v; __syncthreads();
  for (int s = 32; s > 0; s >>= 1) { if (d < s) red[d] += red[d + s]; __syncthreads(); }
  out[(size_t)n * DIM + d] = v * rsqrtf(red[0]);
}

// =====================================================================
// 2. Fused cosine-similarity GEMM (fp32 WMMA) + streaming top-20
//    One wave per 16-row tile; streams all 625 column tiles.
//    Software-pipelined: B fragments for tile ct+1 are issued as one
//    load clause before the WMMA chain for tile ct, so the 16 chained
//    v_wmma_f32_16x16x4_f32 + the top-k insertion hide the L2 latency
//    of the next tile's loads (no per-WMMA s_wait_loadcnt).
// =====================================================================
__global__ void cos_topk_kernel(const float* __restrict__ normed, int* __restrict__ topk_idx) {
  int row0 = blockIdx.x * 16;
  int lane = threadIdx.x;                         // 32 threads = 1 wave
  int m    = lane & 15;
  int koff = (lane >> 4) << 1;                    // 0 or 2

  // A fragments (16x4 f32): lanes 0-15 -> M rows, pair {K,K+1}
  v2f a[16];
  const float* arow = normed + (size_t)(row0 + m) * DIM;
  #pragma unroll
  for (int f = 0; f < 16; ++f) { a[f][0] = arow[4*f + koff]; a[f][1] = arow[4*f + koff + 1]; }

  __shared__ float tv[16][TOPK];
  __shared__ int   ti[16][TOPK];
  __shared__ float sc[16][16];
  if (lane < 16)
    for (int j = 0; j < TOPK; ++j) { tv[lane][j] = -3e38f; ti[lane][j] = 0; }

  const int NT = NN / 16;
  v2f bcur[16], bnxt[16];
  {
    const float* brow = normed + (size_t)m * DIM; // tile 0
    #pragma unroll
    for (int f = 0; f < 16; ++f) { bcur[f][0] = brow[4*f + koff]; bcur[f][1] = brow[4*f + koff + 1]; }
  }
  __syncthreads();

  for (int ct = 0; ct < NT; ++ct) {
    // ---- issue next tile's B loads (single clause, consumed next iter) ----
    if (ct + 1 < NT) {
      const float* brow = normed + ((size_t)(ct + 1) * 16 + m) * DIM;
      __builtin_prefetch(brow + 32 * DIM, 0, 1);  // global_prefetch_b8, 2 tiles ahead
      #pragma unroll
      for (int f = 0; f < 16; ++f) { bnxt[f][0] = brow[4*f + koff]; bnxt[f][1] = brow[4*f + koff + 1]; }
    }

    // ---- 16 chained WMMAs (K = 64) on the current tile ----
    v8f c = {0.f,0.f,0.f,0.f,0.f,0.f,0.f,0.f};
    #pragma unroll
    for (int f = 0; f < 16; ++f)
      c = __builtin_amdgcn_wmma_f32_16x16x4_f32(false, a[f], false, bcur[f], (short)0, c, false, false);

    // ---- transpose 16x16 scores through LDS, then per-row top-k insert ----
    int col0 = ct * 16;
    int n = lane & 15;
    int rbase = (lane < 16) ? 0 : 8;
    #pragma unroll
    for (int r = 0; r < 8; ++r) sc[rbase + r][n] = c[r];
    __syncthreads();

    if (lane < 16) {                              // lane owns row `lane`
      for (int j = 0; j < 16; ++j) {
        float v = sc[lane][j];
        if (v > tv[lane][TOPK - 1]) {
          int p = TOPK - 1;
          while (p > 0 && tv[lane][p - 1] < v) {  // strict: stable for ties (low idx first)
            tv[lane][p] = tv[lane][p - 1]; ti[lane][p] = ti[lane][p - 1]; --p;
          }
          tv[lane][p] = v; ti[lane][p] = col0 + j;
        }
      }
    }
    __syncthreads();

    // ---- rotate double buffer ----
    #pragma unroll
    for (int f = 0; f < 16; ++f) bcur[f] = bnxt[f];
  }

  if (lane < 16)
    for (int j = 0; j < TOPK; ++j)
      topk_idx[(size_t)(row0 + lane) * TOPK + j] = ti[lane][j];
}

// =====================================================================
// 3. GNN: h = x @ W ; s_i = h@ai + emb@aei ; s_j = h@aj + emb@aej
//    (flip=1 reverses the feature axis for the backward branch)
// =====================================================================
__global__ void gnn_h_kernel(const float* __restrict__ x, const float* __restrict__ W,
                             const float* __restrict__ ai, const float* __restrict__ aj,
                             const float* __restrict__ aei, const float* __restrict__ aej,
                             const float* __restrict__ emb,
                             float* __restrict__ h, float* __restrict__ si, float* __restrict__ sj,
                             int flip) {
  int g = blockIdx.x, d = threadIdx.x;            // 64 threads
  const float* xr = x + (size_t)g * FIN;
  float acc = 0.f;
  #pragma unroll
  for (int k = 0; k < FIN; ++k) {
    float xv = flip ? xr[FIN - 1 - k] : xr[k];
    acc += xv * W[k * DIM + d];
  }
  h[(size_t)g * DIM + d] = acc;
  int n = g % NN;
  float e = emb[(size_t)n * DIM + d];
  __shared__ float red[64];
  red[d] = acc * ai[d] + e * aei[d]; __syncthreads();
  for (int s = 32; s > 0; s >>= 1) { if (d < s) red[d] += red[d + s]; __syncthreads(); }
  if (d == 0) si[g] = red[0];
  __syncthreads();
  red[d] = acc * aj[d] + e * aej[d]; __syncthreads();
  for (int s = 32; s > 0; s >>= 1) { if (d < s) red[d] += red[d + s]; __syncthreads(); }
  if (d == 0) sj[g] = red[0];
}

// =====================================================================
// 4. GNN attention aggregation: one wave per node, 20 neighbors.
// =====================================================================
__global__ void gnn_agg_kernel(const float* __restrict__ h, const float* __restrict__ si,
                               const float* __restrict__ sj, const int* __restrict__ topk,
                               const float* __restrict__ bias, float* __restrict__ out) {
  int g = blockIdx.x, lane = threadIdx.x;         // 32 threads
  int b = g / NN, n = g % NN;
  __shared__ float red[32];
  __shared__ float alpha[TOPK];
  __shared__ int   srcg[TOPK];
  float logit = -1e30f;
  if (lane < TOPK) {
    int s  = topk[(size_t)n * TOPK + lane];
    int sg = b * NN + s;
    srcg[lane] = sg;
    float l = si[g] + sj[sg];
    logit = (l > 0.f) ? l : 0.2f * l;             // leaky_relu 0.2
  }
  red[lane] = logit; __syncthreads();
  for (int s = 16; s > 0; s >>= 1) { if (lane < s) red[lane] = fmaxf(red[lane], red[lane + s]); __syncthreads(); }
  float mx = red[0]; __syncthreads();
  float e = (lane < TOPK) ? __expf(logit - mx) : 0.f;
  red[lane] = e; __syncthreads();
  for (int s = 16; s > 0; s >>= 1) { if (lane < s) red[lane] += red[lane + s]; __syncthreads(); }
  float z = red[0];
  if (lane < TOPK) alpha[lane] = e / z;
  __syncthreads();
  for (int d = lane; d < DIM; d += 32) {
    float acc = bias[d];
    for (int j = 0; j < TOPK; ++j) acc += alpha[j] * h[(size_t)srcg[j] * DIM + d];
    out[(size_t)g * DIM + d] = acc;
  }
}

// =====================================================================
// 5. BatchNorm statistics (deterministic column tree-reduction)
// =====================================================================
__global__ void bn_stats_kernel(const float* __restrict__ out, float* __restrict__ mu,
                                float* __restrict__ var) {
  int d = blockIdx.x, t = threadIdx.x;            // 64 blocks x 256
  __shared__ float s1[256], s2[256];
  float a = 0.f, b = 0.f;
  for (int g = t; g < BB * NN; g += 256) {
    float v = out[(size_t)g * DIM + d]; a += v; b += v * v;
  }
  s1[t] = a; s2[t] = b; __syncthreads();
  for (int s = 128; s > 0; s >>= 1) { if (t < s) { s1[t] += s1[t + s]; s2[t] += s2[t + s]; } __syncthreads(); }
  if (t == 0) { float m = s1[0] / (BB * NN); mu[d] = m; var[d] = s2[0] / (BB * NN) - m * m; }
}

// =====================================================================
// 6. BN-apply + ReLU + fusion concat (leaky on relu output == relu)
// =====================================================================
__global__ void fuse_kernel(const float* __restrict__ of, const float* __restrict__ mf,
                            const float* __restrict__ vf, const float* __restrict__ gf_,
                            const float* __restrict__ bf_,
                            const float* __restrict__ ob, const float* __restrict__ mb,
                            const float* __restrict__ vb, const float* __restrict__ gb_,
                            const float* __restrict__ bb_, float* __restrict__ node_h) {
  size_t i = (size_t)blockIdx.x * 256 + threadIdx.x;
  if (i >= (size_t)BB * NN * DIM) return;
  size_t g = i / DIM; int d = (int)(i % DIM);
  float yf = (of[i] - mf[d]) * rsqrtf(vf[d] + EPSF) * gf_[d] + bf_[d];
  float yb = (ob[i] - mb[d]) * rsqrtf(vb[d] + EPSF) * gb_[d] + bb_[d];
  node_h[g * 2 * DIM + d]        = fmaxf(yf, 0.f);
  node_h[g * 2 * DIM + DIM + d]  = fmaxf(yb, 0.f);
}

// =====================================================================
// 7. Column sum of the "ones" half of emb1_w
// =====================================================================
__global__ void colsum_kernel(const float* __restrict__ emb1_w, float* __restrict__ colsum) {
  int d = blockIdx.x, t = threadIdx.x;            // 128 blocks x 256
  __shared__ float s1[256];
  float a = 0.f;
  for (int n = t; n < NN; n += 256) a += emb1_w[(size_t)(NN + n) * DMODEL + d];
  s1[t] = a; __syncthreads();
  for (int s = 128; s > 0; s >>= 1) { if (t < s) s1[t] += s1[t + s]; __syncthreads(); }
  if (t == 0) colsum[d] = s1[0];
}

// =====================================================================
// 8. Token embedding: h0 = X^T @ emb1_w[:N] + colsum + bias + sinusoid
// =====================================================================
__global__ void embed_kernel(const float* __restrict__ x, const float* __restrict__ emb1_w,
                             const float* __restrict__ emb1_b, const float* __restrict__ colsum,
                             float* __restrict__ hio) {
  int bt = blockIdx.x; int b = bt / FIN, t = bt % FIN; int d = threadIdx.x; // 128 threads
  const float* xb = x + (size_t)b * NN * FIN + t;
  float acc = 0.f;
  for (int n = 0; n < NN; ++n) acc += xb[(size_t)n * FIN] * emb1_w[(size_t)n * DMODEL + d];
  float expo  = (float)(2 * (d >> 1)) / (float)DMODEL;
  float angle = (float)t * __expf(-expo * __logf(10000.f));
  float pe    = (d & 1) ? __cosf(angle) : __sinf(angle);
  hio[(size_t)bt * DMODEL + d] = acc + colsum[d] + emb1_b[d] + pe;
}

// =====================================================================
// 9. Full pre-LN encoder layer, one workgroup per batch element
// =====================================================================
__global__ void encoder_kernel(float* __restrict__ hio,
  const float* __restrict__ g1, const float* __restrict__ b1,
  const float* __restrict__ wq, const float* __restrict__ wk, const float* __restrict__ wv,
  const float* __restrict__ fcw, const float* __restrict__ fcb,
  const float* __restrict__ g2, const float* __restrict__ b2,
  const float* __restrict__ w1, const float* __restrict__ bb1,
  const float* __restrict__ w2, const float* __restrict__ bb2) {
  int b = blockIdx.x, tid = threadIdx.x;          // 128 threads
  __shared__ float xb[FIN][DMODEL], xn[FIN][DMODEL];
  __shared__ float qq[FIN][DMODEL], kk[FIN][DMODEL], vv[FIN][DMODEL], cx[FIN][DMODEL];
  __shared__ float ffh[FIN][DINNER];
  __shared__ float sc[2 * FIN * FIN];
  __shared__ float red[DMODEL];
  float* hb = hio + (size_t)b * FIN * DMODEL;
  for (int t = 0; t < FIN; ++t) xb[t][tid] = hb[t * DMODEL + tid];
  __syncthreads();
  // --- LN1 ---
  for (int t = 0; t < FIN; ++t) {
    red[tid] = xb[t][tid]; __syncthreads();
    for (int s = 64; s > 0; s >>= 1) { if (tid < s) red[tid] += red[tid + s]; __syncthreads(); }
    float m = red[0] / DMODEL; __syncthreads();
    float dv = xb[t][tid] - m;
    red[tid] = dv * dv; __syncthreads();
    for (int s = 64; s > 0; s >>= 1) { if (tid < s) red[tid] += red[tid + s]; __syncthreads(); }
    float var = red[0] / DMODEL; __syncthreads();
    xn[t][tid] = dv * rsqrtf(var + EPSF) * g1[tid] + b1[tid];
  }
  __syncthreads();
  // --- QKV ---
  for (int idx = tid; idx < FIN * DMODEL; idx += 128) {
    int t = idx / DMODEL, e = idx % DMODEL;
    float aq = 0.f, ak = 0.f, av = 0.f;
    for (int c = 0; c < DMODEL; ++c) {
      float xv = xn[t][c];
      aq += xv * wq[c * DMODEL + e]; ak += xv * wk[c * DMODEL + e]; av += xv * wv[c * DMODEL + e];
    }
    qq[t][e] = aq; kk[t][e] = ak; vv[t][e] = av;
  }
  __syncthreads();
  // --- scores (2 heads, dk=64), diagonal masked ---
  for (int idx = tid; idx < 2 * FIN * FIN; idx += 128) {
    int hh = idx / (FIN * FIN), r = (idx / FIN) % FIN, c = idx % FIN;
    float a = 0.f;
    for (int d2 = 0; d2 < 64; ++d2) a += qq[r][hh * 64 + d2] * kk[c][hh * 64 + d2];
    a *= 0.125f;                                  // 1/sqrt(64)
    if (r == c) a = -1e9f;
    sc[idx] = a;
  }
  __syncthreads();
  // --- softmax per (head,row) ---
  if (tid < 2 * FIN) {
    float* row = sc + tid * FIN;
    float m = row[0]; for (int c = 1; c < FIN; ++c) m = fmaxf(m, row[c]);
    float z = 0.f;
    for (int c = 0; c < FIN; ++c) { row[c] = __expf(row[c] - m); z += row[c]; }
    float iz = 1.f / z;
    for (int c = 0; c < FIN; ++c) row[c] *= iz;
  }
  __syncthreads();
  // --- ctx = attn @ v ---
  for (int idx = tid; idx < FIN * DMODEL; idx += 128) {
    int t = idx / DMODEL, e = idx % DMODEL, hh = e / 64;
    float a = 0.f;
    const float* row = sc + (hh * FIN + t) * FIN;
    for (int c = 0; c < FIN; ++c) a += row[c] * vv[c][e];
    cx[t][e] = a;
  }
  __syncthreads();
  // --- out proj + residual ---
  for (int idx = tid; idx < FIN * DMODEL; idx += 128) {
    int t = idx / DMODEL, e = idx % DMODEL;
    float a = fcb[e];
    for (int c = 0; c < DMODEL; ++c) a += cx[t][c] * fcw[c * DMODEL + e];
    xb[t][e] += a;
  }
  __syncthreads();
  // --- LN2 ---
  for (int t = 0; t < FIN; ++t) {
    red[tid] = xb[t][tid]; __syncthreads();
    for (int s = 64; s > 0; s >>= 1) { if (tid < s) red[tid] += red[tid + s]; __syncthreads(); }
    float m = red[0] / DMODEL; __syncthreads();
    float dv = xb[t][tid] - m;
    red[tid] = dv * dv; __syncthreads();
    for (int s = 64; s > 0; s >>= 1) { if (tid < s) red[tid] += red[tid + s]; __syncthreads(); }
    float var = red[0] / DMODEL; __syncthreads();
    xn[t][tid] = dv * rsqrtf(var + EPSF) * g2[tid] + b2[tid];
  }
  __syncthreads();
  // --- FFN ---
  for (int idx = tid; idx < FIN * DINNER; idx += 128) {
    int t = idx / DINNER, j = idx % DINNER;
    float a = bb1[j];
    for (int c = 0; c < DMODEL; ++c) a += xn[t][c] * w1[c * DINNER + j];
    ffh[t][j] = fmaxf(a, 0.f);
  }
  __syncthreads();
  for (int idx = tid; idx < FIN * DMODEL; idx += 128) {
    int t = idx / DMODEL, e = idx % DMODEL;
    float a = bb2[e];
    for (int j = 0; j < DINNER; ++j) a += ffh[t][j] * w2[j * DMODEL + e];
    xb[t][e] += a;
  }
  __syncthreads();
  for (int t = 0; t < FIN; ++t) hb[t * DMODEL + tid] = xb[t][tid];
}

// =====================================================================
// 10. Classifier K/V projection of time tokens
// =====================================================================
__global__ void kv_kernel(const float* __restrict__ hio, const float* __restrict__ in_w,
                          const float* __restrict__ in_b, float* __restrict__ kbuf,
                          float* __restrict__ vbuf) {
  int bt = blockIdx.x, d = threadIdx.x;           // 80 blocks x 128
  __shared__ float xs[DMODEL];
  xs[d] = hio[(size_t)bt * DMODEL + d]; __syncthreads();
  float ak = in_b[DMODEL + d], av = in_b[2 * DMODEL + d];
  for (int c = 0; c < DMODEL; ++c) {
    float xv = xs[c];
    ak += xv * in_w[(size_t)(DMODEL + d) * DMODEL + c];
    av += xv * in_w[(size_t)(2 * DMODEL + d) * DMODEL + c];
  }
  kbuf[(size_t)bt * DMODEL + d] = ak;
  vbuf[(size_t)bt * DMODEL + d] = av;
}

// =====================================================================
// 11. Classifier: q-projection + softmax, reduced to Σ_n attn weights.
//     (mean over nodes commutes through ctx @ out_w.T, so only the
//      per-(b,h,t) sum of softmax weights is needed.)
// =====================================================================
__global__ void attw_part_kernel(const float* __restrict__ node_h, const float* __restrict__ kbuf,
                                 const float* __restrict__ in_w, const float* __restrict__ in_b,
                                 float* __restrict__ part) {
  int pb = blockIdx.x, b = blockIdx.y, tid = threadIdx.x;  // 128 threads
  __shared__ float xs[DMODEL], qs[DMODEL], ss[4 * FIN], acc[4 * FIN];
  if (tid < 40) acc[tid] = 0.f;
  __syncthreads();
  int n0 = pb * NODES_PER_BLK;
  for (int nn = 0; nn < NODES_PER_BLK; ++nn) {
    size_t g = (size_t)b * NN + n0 + nn;
    xs[tid] = node_h[g * DMODEL + tid]; __syncthreads();
    float a = in_b[tid];
    for (int c = 0; c < DMODEL; ++c) a += xs[c] * in_w[(size_t)tid * DMODEL + c];
    qs[tid] = a; __syncthreads();
    if (tid < 40) {                               // 4 heads x 10 time tokens
      int hh = tid / FIN, t = tid % FIN;
      const float* kr = kbuf + ((size_t)b * FIN + t) * DMODEL + hh * 32;
      float s = 0.f;
      for (int d2 = 0; d2 < 32; ++d2) s += qs[hh * 32 + d2] * kr[d2];
      ss[tid] = s * 0.1767766953f;                // 1/sqrt(32)
    }
    __syncthreads();
    if (tid < 4) {                                // softmax per head, accumulate
      float m = ss[tid * FIN];
      for (int c = 1; c < FIN; ++c) m = fmaxf(m, ss[tid * FIN + c]);
      float e[FIN]; float z = 0.f;
      for (int c = 0; c < FIN; ++c) { e[c] = __expf(ss[tid * FIN + c] - m); z += e[c]; }
      float iz = 1.f / z;
      for (int c = 0; c < FIN; ++c) acc[tid * FIN + c] += e[c] * iz;
    }
    __syncthreads();
  }
  if (tid < 40) part[((size_t)b * PBLK + pb) * 40 + tid] = acc[tid];
}

__global__ void attw_reduce_kernel(const float* __restrict__ part, float* __restrict__ attw) {
  int i = threadIdx.x;                            // 320 threads
  int b = i / 40, r = i % 40;
  float a = 0.f;
  for (int p = 0; p < PBLK; ++p) a += part[((size_t)b * PBLK + p) * 40 + r];
  attw[i] = a;
}

// =====================================================================
// 12. Final: ctx-mean -> out proj -> classifier head
// =====================================================================
__global__ void final_kernel(const float* __restrict__ attw, const float* __restrict__ vbuf,
                             const float* __restrict__ out_w, const float* __restrict__ out_b,
                             const float* __restrict__ cls_w, const float* __restrict__ cls_b,
                             float* __restrict__ out) {
  int tid = threadIdx.x;                          // 128 threads, 1 block
  __shared__ float cm[DMODEL], mg[DMODEL];
  for (int b = 0; b < BB; ++b) {
    int hh = tid / 32;
    float a = 0.f;
    for (int t = 0; t < FIN; ++t)
      a += attw[b * 40 + hh * FIN + t] * vbuf[((size_t)b * FIN + t) * DMODEL + tid];
    cm[tid] = a * (1.f / NN);
    __syncthreads();
    float m = out_b[tid];
    for (int c = 0; c < DMODEL; ++c) m += cm[c] * out_w[(size_t)tid * DMODEL + c];
    mg[tid] = m; __syncthreads();
    if (tid < 7) {
      float l = cls_b[tid];
      for (int e = 0; e < DMODEL; ++e) l += mg[e] * cls_w[e * 7 + tid];
      out[b * 7 + tid] = l;
    }
    __syncthreads();
  }
}

// =====================================================================
// Host side
// =====================================================================
extern "C" void kernel_launch(void* const* d_in, const int* in_sizes, int n_in,
                              void* d_out, int out_size, void* d_ws, size_t ws_size,
                              hipStream_t stream) {
  const float* data      = (const float*)d_in[0];
  // d_in[1] org_edge_index: unused by reference forward path
  const float* emb_a     = (const float*)d_in[2];
  const float* emb_b     = (const float*)d_in[3];
  const float* gnn_w_f   = (const float*)d_in[4];
  const float* att_i_f   = (const float*)d_in[5];
  const float* att_j_f   = (const float*)d_in[6];
  const float* att_em_i_f= (const float*)d_in[7];
  const float* att_em_j_f= (const float*)d_in[8];
  const float* gnn_bias_f= (const float*)d_in[9];
  const float* bn_g_f    = (const float*)d_in[10];
  const float* bn_b_f    = (const float*)d_in[11];
  const float* gnn_w_b   = (const float*)d_in[12];
  const float* att_i_b   = (const float*)d_in[13];
  const float* att_j_b   = (const float*)d_in[14];
  const float* att_em_i_b= (const float*)d_in[15];
  const float* att_em_j_b= (const float*)d_in[16];
  const float* gnn_bias_b= (const float*)d_in[17];
  const float* bn_g_b    = (const float*)d_in[18];
  const float* bn_b_b    = (const float*)d_in[19];
  const float* emb1_w    = (const float*)d_in[20];
  const float* emb1_b    = (const float*)d_in[21];
  const float* ln1_g     = (const float*)d_in[22];
  const float* ln1_b     = (const float*)d_in[23];
  const float* wq        = (const float*)d_in[24];
  const float* wk        = (const float*)d_in[25];
  const float* wv        = (const float*)d_in[26];
  const float* fc_w      = (const float*)d_in[27];
  const float* fc_b      = (const float*)d_in[28];
  const float* ln2_g     = (const float*)d_in[29];
  const float* ln2_b     = (const float*)d_in[30];
  const float* ff1_w     = (const float*)d_in[31];
  const float* ff1_b     = (const float*)d_in[32];
  const float* ff2_w     = (const float*)d_in[33];
  const float* ff2_b     = (const float*)d_in[34];
  const float* mha_in_w  = (const float*)d_in[35];
  const float* mha_in_b  = (const float*)d_in[36];
  const float* mha_out_w = (const float*)d_in[37];
  const float* mha_out_b = (const float*)d_in[38];
  const float* cls_w     = (const float*)d_in[39];
  const float* cls_b     = (const float*)d_in[40];

  float* ws = (float*)d_ws;
  size_t off = 0;
  float* normed_a = ws + off; off += (size_t)NN * DIM;
  float* normed_b = ws + off; off += (size_t)NN * DIM;
  int*   topk_a   = (int*)(ws + off); off += (size_t)NN * TOPK;
  int*   topk_b   = (int*)(ws + off); off += (size_t)NN * TOPK;
  size_t hoff = off;
  float* h_f      = ws + off; off += (size_t)BB * NN * DIM;
  float* h_b      = ws + off; off += (size_t)BB * NN * DIM;
  float* node_h   = ws + hoff;                     // aliases h_f..h_b (dead by then)
  float* si_f     = ws + off; off += (size_t)BB * NN;
  float* sj_f     = ws + off; off += (size_t)BB * NN;
  float* si_b     = ws + off; off += (size_t)BB * NN;
  float* sj_b     = ws + off; off += (size_t)BB * NN;
  float* out_f    = ws + off; off += (size_t)BB * NN * DIM;
  float* out_b    = ws + off; off += (size_t)BB * NN * DIM;
  float* mu_f     = ws + off; off += DIM;
  float* var_f    = ws + off; off += DIM;
  float* mu_b     = ws + off; off += DIM;
  float* var_b    = ws + off; off += DIM;
  float* colsum   = ws + off; off += DMODEL;
  float* hio      = ws + off; off += (size_t)BB * FIN * DMODEL;
  float* kbuf     = ws + off; off += (size_t)BB * FIN * DMODEL;
  float* vbuf     = ws + off; off += (size_t)BB * FIN * DMODEL;
  float* part     = ws + off; off += (size_t)BB * PBLK * 40;
  float* attw     = ws + off; off += (size_t)BB * 40;
  (void)in_sizes; (void)n_in; (void)out_size; (void)ws_size;

  // --- graph branch: normalize -> WMMA cos + top-k ---
  normalize_kernel<<<NN, 64, 0, stream>>>(emb_a, normed_a);
  normalize_kernel<<<NN, 64, 0, stream>>>(emb_b, normed_b);
  cos_topk_kernel<<<NN / 16, 32, 0, stream>>>(normed_a, topk_a);
  cos_topk_kernel<<<NN / 16, 32, 0, stream>>>(normed_b, topk_b);

  // --- GNN forward + backward branches ---
  gnn_h_kernel<<<BB * NN, 64, 0, stream>>>(data, gnn_w_f, att_i_f, att_j_f, att_em_i_f, att_em_j_f,
                                           emb_a, h_f, si_f, sj_f, 0);
  gnn_h_kernel<<<BB * NN, 64, 0, stream>>>(data, gnn_w_b, att_i_b, att_j_b, att_em_i_b, att_em_j_b,
                                           emb_b, h_b, si_b, sj_b, 1);
  gnn_agg_kernel<<<BB * NN, 32, 0, stream>>>(h_f, si_f, sj_f, topk_a, gnn_bias_f, out_f);
  gnn_agg_kernel<<<BB * NN, 32, 0, stream>>>(h_b, si_b, sj_b, topk_b, gnn_bias_b, out_b);
  bn_stats_kernel<<<DIM, 256, 0, stream>>>(out_f, mu_f, var_f);
  bn_stats_kernel<<<DIM, 256, 0, stream>>>(out_b, mu_b, var_b);
  fuse_kernel<<<(unsigned)(((size_t)BB * NN * DIM + 255) / 256), 256, 0, stream>>>(
      out_f, mu_f, var_f, bn_g_f, bn_b_f, out_b, mu_b, var_b, bn_g_b, bn_b_b, node_h);

  // --- transformer over time tokens ---
  colsum_kernel<<<DMODEL, 256, 0, stream>>>(emb1_w, colsum);
  embed_kernel<<<BB * FIN, 128, 0, stream>>>(data, emb1_w, emb1_b, colsum, hio);
  for (int l = 0; l < 3; ++l)
    for (int r = 0; r < 3; ++r)
      encoder_kernel<<<BB, 128, 0, stream>>>(hio,
          ln1_g + l * DMODEL, ln1_b + l * DMODEL,
          wq + (size_t)l * DMODEL * DMODEL, wk + (size_t)l * DMODEL * DMODEL,
          wv + (size_t)l * DMODEL * DMODEL,
          fc_w + (size_t)l * DMODEL * DMODEL, fc_b + l * DMODEL,
          ln2_g + l * DMODEL, ln2_b + l * DMODEL,
          ff1_w + (size_t)l * DMODEL * DINNER, ff1_b + l * DINNER,
          ff2_w + (size_t)l * DINNER * DMODEL, ff2_b + l * DMODEL);

  // --- classifier ---
  kv_kernel<<<BB * FIN, 128, 0, stream>>>(hio, mha_in_w, mha_in_b, kbuf, vbuf);
  attw_part_kernel<<<dim3(PBLK, BB), 128, 0, stream>>>(node_h, kbuf, mha_in_w, mha_in_b, part);
  attw_reduce_kernel<<<1, 320, 0, stream>>>(part, attw);
  final_kernel<<<1, 128, 0, stream>>>(attw, vbuf, mha_out_w, mha_out_b, cls_w, cls_b, (float*)d_out);
}